// LightGCN_61993557950655
// MI455X (gfx1250) — compile-verified
//
#include <hip/hip_runtime.h>
#include <hip/hip_bf16.h>

// LightGCN: 4x COO SpMM propagation. Memory/atomic bound; whole working set
// (~134MB) fits MI455X's 192MB L2. Strategy: scalar-path (SMEM) edge metadata,
// coalesced 256B row gathers (wave32 x float2), hardware f32 global atomics,
// global_prefetch (gfx1250) to stream the 57.6MB edge arrays from HBM.

#define DIM 64
#define WAVES_PER_BLOCK 8
#define EDGES_PER_WAVE 4
#define THREADS (WAVES_PER_BLOCK * 32)
#define EDGES_PER_BLOCK (WAVES_PER_BLOCK * EDGES_PER_WAVE)

__device__ __forceinline__ void atomic_add_f32(float* p, float v) {
    // HIP's unsafeAtomicAdd lowers to the native global_atomic_add_f32
    // (no-return form -> STOREcnt) instead of a CAS loop.
    unsafeAtomicAdd(p, v);
}

// Generic layer: out[row[e]] += val[e] * x[col[e]], x is [n_nodes, 64] f32.
__global__ __launch_bounds__(THREADS) void spmm_layer(
    const float* __restrict__ x,
    const float* __restrict__ val,
    const int* __restrict__ row,
    const int* __restrict__ col,
    float* __restrict__ out,
    int nedges)
{
    const int lane = threadIdx.x & 31;
    int gwave = blockIdx.x * WAVES_PER_BLOCK + (threadIdx.x >> 5);
    // Force wave-uniform edge base so metadata loads go down the scalar
    // (s_load) path and leave the vector pipe to gathers/atomics.
    int e0 = __builtin_amdgcn_readfirstlane(gwave) * EDGES_PER_WAVE;
    if (e0 >= nedges) return;

    // gfx1250 global_prefetch: stream edge metadata ~16K edges ahead into L2.
    // One wave in 8 issues it (covers a 128B line per array per 32 edges).
    if (((e0 & 31) == 0) && (e0 + 16384) < nedges) {
        __builtin_prefetch(val + e0 + 16384);
        __builtin_prefetch(row + e0 + 16384);
        __builtin_prefetch(col + e0 + 16384);
    }

    float v[EDGES_PER_WAVE];
    int   r[EDGES_PER_WAVE];
    int   c[EDGES_PER_WAVE];
#pragma unroll
    for (int k = 0; k < EDGES_PER_WAVE; ++k) {
        int e = e0 + k;
        bool ok = (e < nedges);       // uniform; tail edges add 0.0f
        int ee = ok ? e : e0;
        v[k] = ok ? val[ee] : 0.0f;
        r[k] = row[ee];
        c[k] = col[ee];
    }

#pragma unroll
    for (int k = 0; k < EDGES_PER_WAVE; ++k) {
        const float2* src =
            reinterpret_cast<const float2*>(x + (size_t)c[k] * DIM);
        float2 xv = src[lane];        // wave loads one contiguous 256B row
        float* dst = out + (size_t)r[k] * DIM + lane * 2;
        atomic_add_f32(dst + 0, v[k] * xv.x);
        atomic_add_f32(dst + 1, v[k] * xv.y);
    }
}

// First layer: gather from the virtual concat(user_emb, item_emb).
__global__ __launch_bounds__(THREADS) void spmm_first(
    const float* __restrict__ user_emb,
    const float* __restrict__ item_emb,
    const float* __restrict__ val,
    const int* __restrict__ row,
    const int* __restrict__ col,
    float* __restrict__ out,
    int nedges, int num_user)
{
    const int lane = threadIdx.x & 31;
    int gwave = blockIdx.x * WAVES_PER_BLOCK + (threadIdx.x >> 5);
    int e0 = __builtin_amdgcn_readfirstlane(gwave) * EDGES_PER_WAVE;
    if (e0 >= nedges) return;

    if (((e0 & 31) == 0) && (e0 + 16384) < nedges) {
        __builtin_prefetch(val + e0 + 16384);
        __builtin_prefetch(row + e0 + 16384);
        __builtin_prefetch(col + e0 + 16384);
    }

    float v[EDGES_PER_WAVE];
    int   r[EDGES_PER_WAVE];
    int   c[EDGES_PER_WAVE];
#pragma unroll
    for (int k = 0; k < EDGES_PER_WAVE; ++k) {
        int e = e0 + k;
        bool ok = (e < nedges);
        int ee = ok ? e : e0;
        v[k] = ok ? val[ee] : 0.0f;
        r[k] = row[ee];
        c[k] = col[ee];
    }

#pragma unroll
    for (int k = 0; k < EDGES_PER_WAVE; ++k) {
        int cc = c[k];                // scalar select between the two halves
        const float* base = (cc < num_user)
            ? (user_emb + (size_t)cc * DIM)
            : (item_emb + (size_t)(cc - num_user) * DIM);
        float2 xv = reinterpret_cast<const float2*>(base)[lane];
        float* dst = out + (size_t)r[k] * DIM + lane * 2;
        atomic_add_f32(dst + 0, v[k] * xv.x);
        atomic_add_f32(dst + 1, v[k] * xv.y);
    }
}

extern "C" void kernel_launch(void* const* d_in, const int* in_sizes, int n_in,
                              void* d_out, int out_size, void* d_ws, size_t ws_size,
                              hipStream_t stream) {
    const float* user_emb = (const float*)d_in[0];
    const float* item_emb = (const float*)d_in[1];
    const float* edge_val = (const float*)d_in[2];
    const int*   edge_row = (const int*)d_in[3];
    const int*   edge_col = (const int*)d_in[4];

    const int num_user = in_sizes[0] / DIM;
    const int num_item = in_sizes[1] / DIM;
    const int n_nodes  = num_user + num_item;
    const int nedges   = in_sizes[2];
    const size_t node_bytes = (size_t)n_nodes * DIM * sizeof(float);

    float* bufA = (float*)d_ws;       // one 38.4MB scratch buffer
    float* outp = (float*)d_out;      // final layer lives here; also ping-pong

    const int nblocks = (nedges + EDGES_PER_BLOCK - 1) / EDGES_PER_BLOCK;

    // Layer 1: concat(user,item) -> bufA
    hipMemsetAsync(bufA, 0, node_bytes, stream);
    spmm_first<<<nblocks, THREADS, 0, stream>>>(
        user_emb, item_emb, edge_val, edge_row, edge_col, bufA, nedges, num_user);

    // Layer 2: bufA -> d_out
    hipMemsetAsync(outp, 0, node_bytes, stream);
    spmm_layer<<<nblocks, THREADS, 0, stream>>>(
        bufA, edge_val, edge_row, edge_col, outp, nedges);

    // Layer 3: d_out -> bufA
    hipMemsetAsync(bufA, 0, node_bytes, stream);
    spmm_layer<<<nblocks, THREADS, 0, stream>>>(
        outp, edge_val, edge_row, edge_col, bufA, nedges);

    // Layer 4: bufA -> d_out   (d_out = [user part ; item part] = full x)
    hipMemsetAsync(outp, 0, node_bytes, stream);
    spmm_layer<<<nblocks, THREADS, 0, stream>>>(
        bufA, edge_val, edge_row, edge_col, outp, nedges);
}